// GCN_21002390077614
// MI455X (gfx1250) — compile-verified
//
#include <hip/hip_runtime.h>
#include <hip/hip_bf16.h>

typedef float v2f __attribute__((ext_vector_type(2)));
typedef float v8f __attribute__((ext_vector_type(8)));

#define DIN 128   // input / hidden feature width (both GEMMs have K=128)
#define DH  128
#define DOUTF 64

// ---------------- utility kernels ----------------

__global__ void k_init_deg(float* __restrict__ deg, int n) {
  int i = blockIdx.x * blockDim.x + threadIdx.x;
  if (i < n) deg[i] = 1.0f;  // self-loop contributes 1 to in-degree
}

__global__ void k_degree(const int* __restrict__ col, float* __restrict__ deg, int e) {
  int i = blockIdx.x * blockDim.x + threadIdx.x;
  if (i < e) unsafeAtomicAdd(&deg[col[i]], 1.0f);
}

__global__ void k_dinv(float* __restrict__ deg, int n) {
  int i = blockIdx.x * blockDim.x + threadIdx.x;
  if (i < n) {
    float d = deg[i];
    deg[i] = (d > 0.0f) ? rsqrtf(d) : 0.0f;   // deg >= 1 always, but match ref
  }
}

// dst[i][j] = bias[j]   (d is a power of two: 128 or 64)
__global__ void k_fill_bias(float* __restrict__ dst, const float* __restrict__ bias,
                            long long total, int d) {
  long long i = (long long)blockIdx.x * blockDim.x + threadIdx.x;
  if (i < total) dst[i] = bias[(int)(i & (d - 1))];
}

__global__ void k_relu(float* __restrict__ x, long long total) {
  long long i = (long long)blockIdx.x * blockDim.x + threadIdx.x;
  if (i < total) x[i] = fmaxf(x[i], 0.0f);
}

// ---------------- fp32 WMMA GEMM:  Y[nrows x DOUT] = X[nrows x 128] @ W[128 x DOUT] ----------------
// One block per 16-row tile. DOUT/16 waves; wave w owns output columns [16w, 16w+16).
// Accumulate over K=128 in steps of 4 with V_WMMA_F32_16X16X4_F32.
template <int DOUT>
__launch_bounds__(DOUT * 2)
__global__ void k_gemm_wmma(const float* __restrict__ X, const float* __restrict__ W,
                            float* __restrict__ Y, int nrows) {
  __shared__ float sx[16 * 132];  // 16 x 128 tile, stride 132 -> conflict-free column reads

  const int tid  = threadIdx.x;
  const int wave = tid >> 5;
  const int lane = tid & 31;
  const int rt   = blockIdx.x * 16;

  // cooperative tile load: 2048 floats
  for (int i = tid; i < 16 * DIN; i += DOUT * 2) {
    int m = i >> 7;              // DIN == 128
    int k = i & (DIN - 1);
    int row = rt + m;
    sx[m * 132 + k] = (row < nrows) ? X[(size_t)row * DIN + k] : 0.0f;
  }
  __syncthreads();

  // fragment index mapping (ISA 7.12.2, 32-bit A 16x4 / B 4x16):
  //   lanes 0-15 hold K = k0+0 (v.x) , k0+1 (v.y)
  //   lanes 16-31 hold K = k0+2 (v.x), k0+3 (v.y)
  const int m    = lane & 15;
  const int hl   = lane >> 4;      // half-wave select
  const int koff = hl * 2;
  const int n    = wave * 16 + (lane & 15);

  v8f acc = {0.f, 0.f, 0.f, 0.f, 0.f, 0.f, 0.f, 0.f};

#pragma unroll
  for (int k0 = 0; k0 < DIN; k0 += 4) {
    v2f a, b;
    a.x = sx[m * 132 + k0 + koff];
    a.y = sx[m * 132 + k0 + koff + 1];
    b.x = W[(size_t)(k0 + koff) * DOUT + n];
    b.y = W[(size_t)(k0 + koff + 1) * DOUT + n];
    acc = __builtin_amdgcn_wmma_f32_16x16x4_f32(false, a, false, b, (short)0, acc,
                                                false, false);
  }

  // D layout: VGPR r -> row (r + 8*half), col = lane&15 within the wave's slice
#pragma unroll
  for (int r = 0; r < 8; ++r) {
    int row = rt + r + hl * 8;
    if (row < nrows) Y[(size_t)row * DOUT + n] = acc[r];
  }
}

// ---------------- edge-parallel scatter-add aggregation ----------------
// One wave per edge; lane holds D/32 contiguous features (vector gather, fp32 atomics).
// Edges [0,nE): (row[e], col[e]); edges [nE, nE+nN): self-loop (i,i).
template <int D>
__global__ void k_aggregate(const float* __restrict__ src,
                            const int* __restrict__ row,
                            const int* __restrict__ col,
                            const float* __restrict__ dinv,
                            float* __restrict__ dst,
                            int nE, int nN) {
  constexpr int PER = D / 32;
  const int lane = threadIdx.x & 31;
  const long long edge =
      (long long)blockIdx.x * (blockDim.x >> 5) + (threadIdx.x >> 5);
  const long long total = (long long)nE + nN;
  if (edge >= total) return;

  int r, c;
  if (edge < nE) {
    r = row[edge];
    c = col[edge];
  } else {
    r = c = (int)(edge - nE);
  }
  const float nrm = dinv[r] * dinv[c];

  const float* s = src + (size_t)r * D + lane * PER;
  float*       d = dst + (size_t)c * D + lane * PER;

  float v[PER];
  if constexpr (PER == 4) {
    const float4 t = *reinterpret_cast<const float4*>(s);
    v[0] = t.x; v[1] = t.y; v[2] = t.z; v[3] = t.w;
  } else {
    const float2 t = *reinterpret_cast<const float2*>(s);
    v[0] = t.x; v[1] = t.y;
  }
#pragma unroll
  for (int j = 0; j < PER; ++j)
    unsafeAtomicAdd(d + j, v[j] * nrm);  // hardware global_atomic_add_f32, L2-resident
}

// ---------------- launcher ----------------

extern "C" void kernel_launch(void* const* d_in, const int* in_sizes, int n_in,
                              void* d_out, int out_size, void* d_ws, size_t ws_size,
                              hipStream_t stream) {
  const float* x  = (const float*)d_in[0];
  const int*   ei = (const int*)d_in[1];
  const float* W1 = (const float*)d_in[2];
  const float* b1 = (const float*)d_in[3];
  const float* W2 = (const float*)d_in[4];
  const float* b2 = (const float*)d_in[5];
  float* out = (float*)d_out;

  const int n = in_sizes[0] / DIN;  // 50000 nodes
  const int e = in_sizes[1] / 2;    // 800000 edges
  const int* row = ei;              // edge_index[0]
  const int* col = ei + e;          // edge_index[1]

  // workspace layout (floats): dinv | xw1 (n*128) | h (n*128) | xw2 (n*64)
  float* ws   = (float*)d_ws;
  float* dinv = ws;
  float* xw1  = ws + (((size_t)n + 255) & ~(size_t)255);
  float* h    = xw1 + (size_t)n * DH;
  float* xw2  = h + (size_t)n * DH;

  const int B = 256;
  const long long nh  = (long long)n * DH;
  const long long no  = (long long)n * DOUTF;
  const long long tot = (long long)e + n;          // edges + self-loops
  const int aggBlocks = (int)((tot + 7) / 8);      // 8 waves (edges) per 256-thread block

  // normalization
  k_init_deg<<<(n + B - 1) / B, B, 0, stream>>>(dinv, n);
  k_degree<<<(e + B - 1) / B, B, 0, stream>>>(col, dinv, e);
  k_dinv<<<(n + B - 1) / B, B, 0, stream>>>(dinv, n);

  // layer 1: h = relu(scatter(xw1) + b1)
  k_gemm_wmma<DH><<<(n + 15) / 16, DH * 2, 0, stream>>>(x, W1, xw1, n);
  k_fill_bias<<<(int)((nh + B - 1) / B), B, 0, stream>>>(h, b1, nh, DH);
  k_aggregate<DH><<<aggBlocks, B, 0, stream>>>(xw1, row, col, dinv, h, e, n);
  k_relu<<<(int)((nh + B - 1) / B), B, 0, stream>>>(h, nh);

  // layer 2: out = scatter(xw2) + b2
  k_gemm_wmma<DOUTF><<<(n + 15) / 16, DOUTF * 2, 0, stream>>>(h, W2, xw2, n);
  k_fill_bias<<<(int)((no + B - 1) / B), B, 0, stream>>>(out, b2, no, DOUTF);
  k_aggregate<DOUTF><<<aggBlocks, B, 0, stream>>>(xw2, row, col, dinv, out, e, n);
}